// PhysicsGNN_54245436949070
// MI455X (gfx1250) — compile-verified
//
#include <hip/hip_runtime.h>

// GCN 2-layer fused pipeline for MI455X (gfx1250, wave32).
// Passes:
//   A) deg init (self-loop = 1)
//   B) deg count over edges (f32 atomics)
//   C) dinv = rsqrt(deg); agg1 = x * dinv^2 (self-loop contribution, non-atomic)
//   D) agg1[dst] += x[src]*norm  (edge scatter, 4 f32 atomics/edge)
//   E) fused per-node matmul: h = relu(agg1@W1+b1) via V_WMMA_F32_16X16X4_F32,
//      p = h@W2 via LDS; out = b2 + p*dinv^2 (self-loop term)
//   F) out[dst] += p[src]*norm  (edge scatter, 2 f32 atomics/edge)

typedef float v2f __attribute__((ext_vector_type(2)));
typedef float v8f __attribute__((ext_vector_type(8)));

#define WPB 4  // waves (16-node WMMA tiles) per block in pass E

__device__ __forceinline__ void atomAddF(float* p, float v) {
    unsafeAtomicAdd(p, v);  // -> global_atomic_add_f32
}

__global__ void k_init_deg(float* __restrict__ deg, int N) {
    int i = blockIdx.x * blockDim.x + threadIdx.x;
    if (i < N) deg[i] = 1.0f;  // self-loop
}

__global__ void k_count_deg(const long long* __restrict__ dst,
                            float* __restrict__ deg, int E) {
    int i = blockIdx.x * blockDim.x + threadIdx.x;
    if (i < E) atomAddF(&deg[(int)dst[i]], 1.0f);
}

// dinv in place of deg; agg1 <- x * dinv^2 (self-loop), plain stores.
__global__ void k_dinv_selfloop(const float4* __restrict__ x4,
                                float* __restrict__ dinv_io,
                                float4* __restrict__ agg1, int N) {
    int i = blockIdx.x * blockDim.x + threadIdx.x;
    if (i >= N) return;
    float r = rsqrtf(dinv_io[i]);  // deg >= 1 always
    dinv_io[i] = r;
    float s = r * r;
    float4 xv = x4[i];
    agg1[i] = make_float4(xv.x * s, xv.y * s, xv.z * s, xv.w * s);
}

__global__ void k_scatter1(const long long* __restrict__ src,
                           const long long* __restrict__ dst,
                           const float4* __restrict__ x4,
                           const float* __restrict__ dinv,
                           float* __restrict__ agg1, int E) {
    int i = blockIdx.x * blockDim.x + threadIdx.x;
    if (i >= E) return;
    int s = (int)src[i], d = (int)dst[i];
    float n = dinv[s] * dinv[d];
    float4 xv = x4[s];
    float* a = agg1 + (size_t)d * 4;
    atomAddF(a + 0, xv.x * n);
    atomAddF(a + 1, xv.y * n);
    atomAddF(a + 2, xv.z * n);
    atomAddF(a + 3, xv.w * n);
}

// Pass E: one wave = 16 nodes. h(16x64) = relu(agg1(16x4) @ W1(4x64) + b1) via
// 4x V_WMMA_F32_16X16X4_F32; stage h in LDS (stride 65); p(16x2) = h @ W2;
// out = b2 + p * dinv^2 (self-loop term of layer 2).
__global__ void __launch_bounds__(32 * WPB)
k_fused_mm(const float* __restrict__ agg1,
           const float* __restrict__ W1, const float* __restrict__ b1,
           const float* __restrict__ W2, const float* __restrict__ b2,
           const float* __restrict__ dinv,
           float* __restrict__ p, float* __restrict__ out,
           int ntiles, int N) {
    __shared__ float lds[WPB][16 * 65];
    const int lane = threadIdx.x & 31;
    const int wv   = threadIdx.x >> 5;
    const int tile = blockIdx.x * WPB + wv;
    const bool active = tile < ntiles;  // wave-uniform
    const int l16  = lane & 15;
    const int half = lane >> 4;
    const int base = tile * 16;

    if (active) {
        // A layout (16x4 f32): lanes 0-15 hold K=0,1; lanes 16-31 hold K=2,3; M = lane%16
        int arow = base + l16;
        if (arow >= N) arow = N - 1;  // clamp keeps EXEC full
        v2f a;
        a.x = agg1[arow * 4 + 2 * half + 0];
        a.y = agg1[arow * 4 + 2 * half + 1];
#pragma unroll
        for (int ct = 0; ct < 4; ++ct) {
            const int col = 16 * ct + l16;
            // B layout (4x16 f32): vgpr v holds row k = v + 2*(lane/16), N = lane%16
            v2f b;
            b.x = W1[(2 * half + 0) * 64 + col];
            b.y = W1[(2 * half + 1) * 64 + col];
            v8f c = {};
            c = __builtin_amdgcn_wmma_f32_16x16x4_f32(
                    false, a, false, b, (short)0, c, false, false);
            // C layout: lane holds column `col`, rows m = v + 8*half
            const float bias = b1[col];
#pragma unroll
            for (int v = 0; v < 8; ++v) {
                float h = c[v] + bias;
                h = fmaxf(h, 0.0f);  // relu
                lds[wv][(v + 8 * half) * 65 + col] = h;
            }
        }
    }
    __syncthreads();
    if (active) {
        const int node = base + l16;
        const int j = half;  // lanes 0-15 -> p[:,0], lanes 16-31 -> p[:,1]
        float sum = 0.0f;
#pragma unroll
        for (int n = 0; n < 64; ++n)
            sum = fmaf(lds[wv][l16 * 65 + n], W2[n * 2 + j], sum);
        if (node < N) {
            p[node * 2 + j] = sum;
            const float r = dinv[node];
            out[node * 2 + j] = b2[j] + sum * (r * r);
        }
    }
}

__global__ void k_scatter2(const long long* __restrict__ src,
                           const long long* __restrict__ dst,
                           const float* __restrict__ p,
                           const float* __restrict__ dinv,
                           float* __restrict__ out, int E) {
    int i = blockIdx.x * blockDim.x + threadIdx.x;
    if (i >= E) return;
    int s = (int)src[i], d = (int)dst[i];
    float n = dinv[s] * dinv[d];
    atomAddF(&out[d * 2 + 0], p[s * 2 + 0] * n);
    atomAddF(&out[d * 2 + 1], p[s * 2 + 1] * n);
}

extern "C" void kernel_launch(void* const* d_in, const int* in_sizes, int n_in,
                              void* d_out, int out_size, void* d_ws, size_t ws_size,
                              hipStream_t stream) {
    const float*     x  = (const float*)d_in[0];
    const long long* ei = (const long long*)d_in[1];  // [2,E] int64
    const float*     W1 = (const float*)d_in[2];      // [4,64]
    const float*     b1 = (const float*)d_in[3];      // [64]
    const float*     W2 = (const float*)d_in[4];      // [64,2]
    const float*     b2 = (const float*)d_in[5];      // [2]
    float* out = (float*)d_out;                       // [N,2]

    const int N = in_sizes[0] / 4;
    const int E = in_sizes[1] / 2;
    const long long* src = ei;
    const long long* dst = ei + E;

    // Workspace: deg/dinv [N] | agg1 [N,4] | p [N,2]  => 7N floats (28 MB)
    float* ws   = (float*)d_ws;
    float* deg  = ws;
    float* agg1 = ws + (size_t)N;
    float* p    = ws + (size_t)N * 5;

    const int TB = 256;
    const int gN = (N + TB - 1) / TB;
    const int gE = (E + TB - 1) / TB;

    k_init_deg<<<gN, TB, 0, stream>>>(deg, N);
    k_count_deg<<<gE, TB, 0, stream>>>(dst, deg, E);
    k_dinv_selfloop<<<gN, TB, 0, stream>>>((const float4*)x, deg, (float4*)agg1, N);
    k_scatter1<<<gE, TB, 0, stream>>>(src, dst, (const float4*)x, deg, agg1, E);

    const int ntiles = (N + 15) / 16;
    const int gF = (ntiles + WPB - 1) / WPB;
    k_fused_mm<<<gF, 32 * WPB, 0, stream>>>(agg1, W1, b1, W2, b2, deg, p, out, ntiles, N);

    k_scatter2<<<gE, TB, 0, stream>>>(src, dst, p, deg, out, E);
}